// SubKnotHasher_56126632624428
// MI455X (gfx1250) — compile-verified
//
#include <hip/hip_runtime.h>
#include <math.h>

// SU(2)-chain eta kernel for MI455X (gfx1250, wave32).
// GEMM part (x · W^T, N=4) on the matrix pipe via V_WMMA_F32_16X16X4_F32;
// kernel is HBM-bound (~11.5us floor for 268MB @ 23.3TB/s), so the design
// goal is maximum memory-level parallelism: low VGPR pressure (no hoisted
// W fragments), deep load clauses, prefetch of the next grid-stride tile.

typedef float v2f __attribute__((ext_vector_type(2)));
typedef float v8f __attribute__((ext_vector_type(8)));

#define DIM    256
#define TILE   16      // tokens per wave-tile (= WMMA N)
#define KSTEPS (DIM/4) // 64 WMMAs of K=4 sweep K=256

struct c2 { float r, i; };
__device__ __forceinline__ c2 cmul(c2 a, c2 b) {
  return { a.r*b.r - a.i*b.i, a.r*b.i + a.i*b.r };
}
__device__ __forceinline__ c2 cadd(c2 a, c2 b) { return { a.r+b.r, a.i+b.i }; }

__global__ __launch_bounds__(256) void su2eta_wmma(
    const float* __restrict__ x, const float* __restrict__ W,
    const float* __restrict__ bias, float* __restrict__ out, int nTiles)
{
  const int lane        = threadIdx.x & 31;
  const int wavesPerBlk = blockDim.x >> 5;
  const int wid         = blockIdx.x * wavesPerBlk + (threadIdx.x >> 5);
  const int nWaves      = gridDim.x * wavesPerBlk;

  // WMMA fragment addressing (wave32, 16x16x4 f32):
  //   A (16x4, M=weight row): lane L holds A[L&15][k0 + 2*(L>>4) + {0,1}]
  //   B (4x16, N=token)     : lane L holds B[k0 + 2*(L>>4) + {0,1}][L&15]
  const int col  = lane & 15;          // token-in-tile / weight-row id
  const int koff = (lane >> 4) << 1;   // 0 or 2

  // Rows 4..15 of A duplicate rows 0..3 (m&3): they only pollute D rows
  // 4..15, which are never read. Keeps the fragment load branchless.
  const float* wrow = W + (size_t)(col & 3) * DIM + koff;

  const float b0 = bias[0], b1 = bias[1], b2 = bias[2], b3 = bias[3];

  for (int tile = wid; tile < nTiles; tile += nWaves) {
    const float* brow = x + (size_t)(tile * TILE + col) * DIM + koff;

    // Defeat LICM on the W-fragment loads WITHOUT erasing the pointer's
    // address space (a laundered pointer degrades to flat_load, which
    // burns DScnt + the LDS pipe). Laundering an integer 0 offset keeps
    // provenance rooted at W (=> global_load_b64) while making the 64
    // A-fragment loads loop-variant, so they are reloaded from L0 per
    // tile (~free, W is 4KB) instead of living in ~128 VGPRs.
    int zero = 0;
    asm volatile("" : "+v"(zero));
    const float* wr = wrow + zero;

    // Keep HBM busy: speculative prefetch of our next grid-stride tile.
    if (tile + nWaves < nTiles)
      __builtin_prefetch(x + (size_t)((tile + nWaves) * TILE + col) * DIM, 0, 1);

    v8f c = {0.f, 0.f, 0.f, 0.f, 0.f, 0.f, 0.f, 0.f};
#pragma unroll
    for (int k = 0; k < KSTEPS; ++k) {
      v2f a  = *(const v2f*)(wr + (k << 2));     // b64, L0 hit (W = 4KB)
      v2f bb = *(const v2f*)(brow + (k << 2));   // b64, line reused 16x via L0
      // 8 args: (neg_a, A, neg_b, B, c_mod, C, reuse_a, reuse_b)
      c = __builtin_amdgcn_wmma_f32_16x16x4_f32(
              false, a, false, bb, (short)0, c, false, false);
    }

    // Lane t (t<16) now holds token t's params in c[0..3] (D rows M=0..3).
    float ar = c[0] + b0, ai = c[1] + b1;
    float br = c[2] + b2, bi = c[3] + b3;

    float inv = 1.0f / sqrtf(ar*ar + ai*ai + br*br + bi*bi + 1e-8f);
    float Ar = ar*inv, Ai = ai*inv, Br = br*inv, Bi = bi*inv;

    // Chain product of 8 SU(2) blocks [[a,b],[-conj(b),conj(a)]] per group.
    // Groups are lanes {0..7} and {8..15}; every lane of a group computes
    // the same product redundantly via shuffles (trivial VALU cost).
    const int src = lane & ~7;
    float mAr = __shfl(Ar, src), mAi = __shfl(Ai, src);
    float mBr = __shfl(Br, src), mBi = __shfl(Bi, src);
    c2 s00{ mAr,  mAi}, s01{mBr,  mBi};
    c2 s10{-mBr,  mBi}, s11{mAr, -mAi};
#pragma unroll
    for (int iB = 1; iB < 8; ++iB) {
      mAr = __shfl(Ar, src + iB); mAi = __shfl(Ai, src + iB);
      mBr = __shfl(Br, src + iB); mBi = __shfl(Bi, src + iB);
      c2 m00{ mAr,  mAi}, m01{mBr,  mBi};
      c2 m10{-mBr,  mBi}, m11{mAr, -mAi};
      c2 n00 = cadd(cmul(s00, m00), cmul(s01, m10));
      c2 n01 = cadd(cmul(s00, m01), cmul(s01, m11));
      c2 n10 = cadd(cmul(s10, m00), cmul(s11, m10));
      c2 n11 = cadd(cmul(s10, m01), cmul(s11, m11));
      s00 = n00; s01 = n01; s10 = n10; s11 = n11;
    }

    c2 det = { s00.r*s11.r - s00.i*s11.i - (s01.r*s10.r - s01.i*s10.i),
               s00.r*s11.i + s00.i*s11.r - (s01.r*s10.i + s01.i*s10.r) };
    float eta = atan2f(det.i, det.r) * 0.3183098861837907f; // 1/pi

    if (lane < 16 && (lane & 7) == 0)
      out[(size_t)tile * 2 + (lane >> 3)] = eta;
  }
}

extern "C" void kernel_launch(void* const* d_in, const int* in_sizes, int n_in,
                              void* d_out, int out_size, void* d_ws, size_t ws_size,
                              hipStream_t stream) {
  (void)n_in; (void)out_size; (void)d_ws; (void)ws_size;
  const float* x    = (const float*)d_in[0];
  const float* W    = (const float*)d_in[1];
  const float* bias = (const float*)d_in[2];
  float* out        = (float*)d_out;

  const int tokens = in_sizes[0] / DIM;   // 262144 for the reference shapes
  const int nTiles = tokens / TILE;       // 16384

  const int wavesPerBlock = 8;            // 256 threads, wave32
  int blocks = (nTiles + wavesPerBlock - 1) / wavesPerBlock;
  if (blocks > 1024) blocks = 1024;       // ~2 tiles/wave: prefetch is live
  if (blocks < 1) blocks = 1;

  su2eta_wmma<<<blocks, 256, 0, stream>>>(x, W, bias, out, nTiles);
}